// CrossConvV2_72189810312037
// MI455X (gfx1250) — compile-verified
//
#include <hip/hip_runtime.h>
#include <math.h>

typedef float v2f __attribute__((ext_vector_type(2)));
typedef float v8f __attribute__((ext_vector_type(8)));

#define DIST       3.0f
#define COEFF      10.0f
#define NEG_SLOPE  0.3f
#define NPTS       1024
#define M_PROBES   26
#define OUTF       6
#define KDIM       78    // m * f
#define KPAD       80    // padded K for WMMA (20 steps of 4)
#define NCOLS      30    // num_convs * out_feats
#define TILE_L     16
#define NTHREADS   512   // 16 waves
#define NWAVES     (NTHREADS / 32)
#define TWO_PI     6.28318530717958647692f

__global__ __launch_bounds__(NTHREADS) void CrossConvV2_kernel(
    const float* __restrict__ points,   // (B, 1024, 6)
    const float* __restrict__ W,        // (5, 78, 6)
    const float* __restrict__ bias,     // (5, 6)
    float* __restrict__ out)            // (B, 1024, 33)
{
    __shared__ float sc[NPTS][4];            // coords (padded, b128-friendly)
    __shared__ float sf[NPTS][4];            // feats pre-scaled by COEFF
    __shared__ float sprobe[M_PROBES][3];
    __shared__ float swf[TILE_L][KPAD];      // reduced weighted feats, K-padded
    __shared__ float sWB[32][KPAD];          // W, col-major, zero-padded

    const int tid   = threadIdx.x;
    const int lane  = tid & 31;
    const int wave  = tid >> 5;              // 0..15
    const int bb    = blockIdx.x >> 6;       // batch (64 tiles of 16 per batch)
    const int ltile = blockIdx.x & 63;

    // ---- probe points (26, once per block) ----
    if (tid < M_PROBES) {
        float a, bang;
        if (tid < 24) {
            int j = tid >> 3, i = tid & 7;
            a    = ((float)j * 0.125f - 0.125f) * TWO_PI;
            bang = ((float)i * 0.125f + (float)(j - 1) * 0.0625f) * TWO_PI;
        } else {
            a    = (tid == 24 ? -0.25f : 0.25f) * TWO_PI;
            bang = 0.0f;
        }
        float sa = sinf(a), ca = cosf(a);
        sprobe[tid][0] = sa * DIST;
        sprobe[tid][1] = ca * cosf(bang) * DIST;
        sprobe[tid][2] = ca * sinf(bang) * DIST;
    }

    // ---- stage batch coords/feats; feats pre-scaled by COEFF ----
    const float* pbase = points + (size_t)bb * NPTS * 6;
    for (int n = tid; n < NPTS; n += NTHREADS) {
        const float* p = pbase + n * 6;
        sc[n][0] = p[0];          sc[n][1] = p[1];          sc[n][2] = p[2];
        sf[n][0] = p[3] * COEFF;  sf[n][1] = p[4] * COEFF;  sf[n][2] = p[5] * COEFF;
    }
    // ---- stage W col-major, zero-padded: sWB[col][k] ----
    for (int i = tid; i < 32 * KPAD; i += NTHREADS) {
        int col = i / KPAD, k = i % KPAD;
        float v = 0.0f;
        if (col < NCOLS && k < KDIM)
            v = W[(size_t)(col / OUTF) * KDIM * OUTF + (size_t)k * OUTF + (col % OUTF)];
        sWB[col][k] = v;
    }
    for (int i = tid; i < TILE_L * KPAD; i += NTHREADS) ((float*)swf)[i] = 0.0f;
    __syncthreads();

    // ---- heavy VALU loop: lanes 0-15 -> point l, lane half -> 13 probes ----
    const int lsub  = lane & 15;
    const int mhalf = lane >> 4;
    const int lglob = ltile * TILE_L + lsub;
    const float clx = sc[lglob][0], cly = sc[lglob][1], clz = sc[lglob][2];

    // per-probe-center constants: -2p and |p|^2
    float qx[13], qy[13], qz[13], pp[13];
#pragma unroll
    for (int mm = 0; mm < 13; ++mm) {
        int m = mhalf * 13 + mm;
        float px = clx + sprobe[m][0];
        float py = cly + sprobe[m][1];
        float pz = clz + sprobe[m][2];
        qx[mm] = -2.0f * px;  qy[mm] = -2.0f * py;  qz[mm] = -2.0f * pz;
        pp[mm] = fmaf(px, px, fmaf(py, py, pz * pz));
    }
    float acc[13][3];
#pragma unroll
    for (int mm = 0; mm < 13; ++mm) { acc[mm][0] = 0.f; acc[mm][1] = 0.f; acc[mm][2] = 0.f; }

    const int n0 = wave * (NPTS / NWAVES);
    for (int n = n0; n < n0 + NPTS / NWAVES; ++n) {
        float cnx = sc[n][0], cny = sc[n][1], cnz = sc[n][2];
        float fx  = sf[n][0], fy  = sf[n][1], fz  = sf[n][2];
        float t0  = fmaf(cnx, cnx, fmaf(cny, cny, fmaf(cnz, cnz, COEFF)));
#pragma unroll
        for (int mm = 0; mm < 13; ++mm) {
            // sq + COEFF = |p|^2 + (|c|^2 + COEFF) - 2 c.p
            float t = pp[mm] + t0;
            t = fmaf(qx[mm], cnx, t);
            t = fmaf(qy[mm], cny, t);
            t = fmaf(qz[mm], cnz, t);
            float r = __builtin_amdgcn_rcpf(t);   // wgt/COEFF; feats carry COEFF
            acc[mm][0] = fmaf(r, fx, acc[mm][0]);
            acc[mm][1] = fmaf(r, fy, acc[mm][1]);
            acc[mm][2] = fmaf(r, fz, acc[mm][2]);
        }
    }

    // ---- cross-wave reduction via LDS float atomics (ds_add_f32) ----
#pragma unroll
    for (int mm = 0; mm < 13; ++mm) {
        int m = mhalf * 13 + mm;
        atomicAdd(&swf[lsub][m * 3 + 0], acc[mm][0]);
        atomicAdd(&swf[lsub][m * 3 + 1], acc[mm][1]);
        atomicAdd(&swf[lsub][m * 3 + 2], acc[mm][2]);
    }
    __syncthreads();

    const float inv_n = 1.0f / (float)NPTS;

    if (wave < 2) {
        // ---- WMMA f32 16x16x4: (16 x 80) @ (80 x 32), padded with zeros ----
        const int ntile = wave;
        const int bcol  = ntile * 16 + lsub;     // this lane's B/D column
        v8f cacc = {0.f, 0.f, 0.f, 0.f, 0.f, 0.f, 0.f, 0.f};
#pragma unroll
        for (int kk = 0; kk < 20; ++kk) {
            int kb = kk * 4 + mhalf * 2;         // lanes 16-31 hold K+2,K+3
            v2f afrag, bfrag;
            afrag.x = swf[lsub][kb]     * inv_n;
            afrag.y = swf[lsub][kb + 1] * inv_n;
            bfrag.x = sWB[bcol][kb];
            bfrag.y = sWB[bcol][kb + 1];
            cacc = __builtin_amdgcn_wmma_f32_16x16x4_f32(
                false, afrag, false, bfrag, (short)0, cacc, false, false);
        }
        if (bcol < NCOLS) {
            int k = bcol / OUTF, o = bcol % OUTF;
            float bv = bias[k * OUTF + o];
#pragma unroll
            for (int v = 0; v < 8; ++v) {        // D VGPR v: M = v + 8*mhalf
                int l = v + mhalf * 8;
                float h = cacc[v] + bv;
                h = (h > 0.f) ? h : NEG_SLOPE * h;
                out[((size_t)(bb * NPTS + ltile * TILE_L + l)) * 33 + 3 + bcol] = h;
            }
        }
    } else if (wave == 2) {
        // ---- coord passthrough: out[..., 0:3] = centers ----
        for (int idx = lane; idx < TILE_L * 3; idx += 32) {
            int pt = idx / 3, c = idx % 3;
            out[((size_t)(bb * NPTS + ltile * TILE_L + pt)) * 33 + c] =
                sc[ltile * TILE_L + pt][c];
        }
    }
}

extern "C" void kernel_launch(void* const* d_in, const int* in_sizes, int n_in,
                              void* d_out, int out_size, void* d_ws, size_t ws_size,
                              hipStream_t stream) {
    const float* points = (const float*)d_in[0];   // (2,1024,6)  f32
    const float* W      = (const float*)d_in[1];   // (5,78,6)    f32
    const float* b      = (const float*)d_in[2];   // (5,6)       f32
    float*       out    = (float*)d_out;           // (2,1024,33) f32

    const int total_pts = in_sizes[0] / 6;         // B * n = 2048
    const int blocks    = total_pts / TILE_L;      // 128
    CrossConvV2_kernel<<<dim3(blocks), dim3(NTHREADS), 0, stream>>>(points, W, b, out);
}